// GCN_44719199486429
// MI455X (gfx1250) — compile-verified
//
#include <hip/hip_runtime.h>

// ---------------------------------------------------------------------------
// GCN layer for MI455X (gfx1250, wave32):
//   agg = segment_sum(feature[src], dst)   (scatter-add, memory bound)
//   out = agg @ W^T + b                    (f32 WMMA 16x16x4, K=128)
// ---------------------------------------------------------------------------

typedef __attribute__((ext_vector_type(2))) float v2f;
typedef __attribute__((ext_vector_type(8))) float v8f;

#define N_NODES 50000
#define FDIM    128
#define EDGES   625000

// ---------------------------------------------------------------------------
// Kernel 1: zero the aggregation scratch (float4 stores, 6250*256*16B = 25.6MB)
// ---------------------------------------------------------------------------
__global__ void gcn_zero_ws(float4* __restrict__ ws) {
    const size_t i = (size_t)blockIdx.x * blockDim.x + threadIdx.x;
    ws[i] = make_float4(0.f, 0.f, 0.f, 0.f);
}

// ---------------------------------------------------------------------------
// Kernel 2: edge scatter-add. One wave32 per edge: lane L loads float4
// feature[src][L*4 .. L*4+3] (coalesced 512B per wave) and atomically adds
// into agg[dst]. agg (25.6MB) is L2-resident (192MB L2) so atomics stay on
// chip; gather reads of `feature` are the dominant HBM traffic (~320MB).
// ---------------------------------------------------------------------------
__global__ void gcn_scatter_edges(const float* __restrict__ feature,
                                  const int*   __restrict__ src,
                                  const int*   __restrict__ dst,
                                  float*       __restrict__ agg) {
    const int wave = threadIdx.x >> 5;            // 8 waves per block
    const int lane = threadIdx.x & 31;
    const int e    = blockIdx.x * 8 + wave;       // EDGES = 78125 * 8 exactly

    const int s = src[e];
    const int d = dst[e];

    const float4 f =
        reinterpret_cast<const float4*>(feature + (size_t)s * FDIM)[lane];

    float* o = agg + (size_t)d * FDIM + lane * 4;
    atomicAdd(o + 0, f.x);
    atomicAdd(o + 1, f.y);
    atomicAdd(o + 2, f.z);
    atomicAdd(o + 3, f.w);
}

// ---------------------------------------------------------------------------
// Kernel 3: out = agg @ W^T + b via V_WMMA_F32_16X16X4_F32.
// Block = 256 threads = 8 waves; block handles 16 output rows, wave w handles
// column tile [w*16, w*16+16). K=128 -> 32 WMMA issues per tile.
//
// Fragment layouts (ISA 7.12.2, f32, wave32):
//   A 16x4 : lane L -> M = L%16 ; VGPR0 = K0, VGPR1 = K0+1, K0 = 2*(L>=16)
//   B 4x16 : lane L -> N = L%16 ; VGPR0 = K0, VGPR1 = K0+1 (symmetric)
//   C 16x16: lane L -> N = L%16 ; VGPR r -> M = r + 8*(L>=16)
// B[k][n] = W[n][k]  (since out = agg * W^T), so A and B loads share the
// same row-major addressing pattern.
// ---------------------------------------------------------------------------
__global__ void gcn_gemm_wmma(const float* __restrict__ agg,
                              const float* __restrict__ W,
                              const float* __restrict__ bias,
                              float*       __restrict__ out) {
    const int wave = threadIdx.x >> 5;     // 0..7 : column tile index
    const int lane = threadIdx.x & 31;
    const int half = lane >> 4;            // 0 or 1
    const int m16  = lane & 15;

    const int row_base = blockIdx.x * 16;  // N_NODES = 3125 * 16 exactly
    const int n_base   = wave * 16;

    const float* __restrict__ arow = agg + (size_t)(row_base + m16) * FDIM + 2 * half;
    const float* __restrict__ brow = W   + (size_t)(n_base   + m16) * FDIM + 2 * half;

    v8f c = {};
#pragma unroll
    for (int k = 0; k < FDIM; k += 4) {
        v2f a, b;
        a.x = arow[k];
        a.y = arow[k + 1];
        b.x = brow[k];
        b.y = brow[k + 1];
        // 8 args: (neg_a, A, neg_b, B, c_mod, C, reuse_a, reuse_b)
        c = __builtin_amdgcn_wmma_f32_16x16x4_f32(
                false, a, false, b, (short)0, c, false, false);
    }

    const float bn = bias[n_base + m16];
#pragma unroll
    for (int r = 0; r < 8; ++r) {
        const int m = r + 8 * half;
        out[(size_t)(row_base + m) * FDIM + n_base + m16] = c[r] + bn;
    }
}

// ---------------------------------------------------------------------------
// Host-side launch (graph-capture safe: only kernel launches on `stream`)
// ---------------------------------------------------------------------------
extern "C" void kernel_launch(void* const* d_in, const int* in_sizes, int n_in,
                              void* d_out, int out_size, void* d_ws, size_t ws_size,
                              hipStream_t stream) {
    const float* feature = (const float*)d_in[0];
    const int*   src     = (const int*)  d_in[1];
    const int*   dst     = (const int*)  d_in[2];
    const float* W       = (const float*)d_in[3];
    const float* bias    = (const float*)d_in[4];

    float* out = (float*)d_out;
    float* agg = (float*)d_ws;   // N_NODES * FDIM floats = 25.6 MB

    // 1) agg = 0
    gcn_zero_ws<<<(N_NODES * FDIM / 4) / 256, 256, 0, stream>>>((float4*)agg);

    // 2) agg[dst[e]] += feature[src[e]]
    gcn_scatter_edges<<<EDGES / 8, 256, 0, stream>>>(feature, src, dst, agg);

    // 3) out = agg @ W^T + bias
    gcn_gemm_wmma<<<N_NODES / 16, 256, 0, stream>>>(agg, W, bias, out);
}